// GMMAttention_81552839017084
// MI455X (gfx1250) — compile-verified
//
#include <hip/hip_runtime.h>

#define B_SZ   512
#define M_MIX  32
#define D_DIM  1024
#define T_MAX  4096
#define N3M    96   // 3*M_MIX

typedef __attribute__((ext_vector_type(2))) float v2f;
typedef __attribute__((ext_vector_type(8))) float v8f;

// ---------------------------------------------------------------------------
// Kernel A: params[512,96] = query[512,1024] @ W[1024,96] + b
// One wave (32 threads) per 16x16 output tile, K stepped by 4 with
// V_WMMA_F32_16X16X4_F32 (full f32, no precision loss).
// grid = (96/16, 512/16) = (6, 32) waves.
// ---------------------------------------------------------------------------
__global__ __launch_bounds__(32)
void gmm_proj_wmma(const float* __restrict__ query,
                   const float* __restrict__ W,
                   const float* __restrict__ bias,
                   float* __restrict__ params)
{
    const int ntile = blockIdx.x;           // 0..5   (N direction, 96/16)
    const int mtile = blockIdx.y;           // 0..31  (M direction, 512/16)
    const int lane  = threadIdx.x;          // 0..31
    const int half  = lane >> 4;            // 0: K0/K1,  1: K2/K3
    const int r     = lane & 15;

    const int arow = mtile * 16 + r;        // A row held by this lane
    const int bcol = ntile * 16 + r;        // B/C column held by this lane

    v8f c = {};
    #pragma unroll 4
    for (int kk = 0; kk < D_DIM; kk += 4) {
        const int k0 = kk + 2 * half;
        v2f a, bm;
        // A 16x4: VGPR0 = K=k0, VGPR1 = K=k0+1 (ISA 7.12.2, 32-bit A layout)
        a.x  = query[arow * D_DIM + k0];
        a.y  = query[arow * D_DIM + k0 + 1];
        // B 4x16: row striped across lanes within a VGPR (symmetric layout)
        bm.x = W[(k0    ) * N3M + bcol];
        bm.y = W[(k0 + 1) * N3M + bcol];
        c = __builtin_amdgcn_wmma_f32_16x16x4_f32(
                /*neg_a=*/false, a, /*neg_b=*/false, bm,
                /*c_mod=*/(short)0, c, /*reuse_a=*/false, /*reuse_b=*/false);
    }

    const float bv = bias[bcol];
    #pragma unroll
    for (int v = 0; v < 8; ++v) {
        // C/D layout: VGPR v -> M = v + 8*half, N = r
        const int row = mtile * 16 + v + 8 * half;
        params[row * N3M + bcol] = c[v] + bv;
    }
}

// ---------------------------------------------------------------------------
// Kernel B: one block per batch row.
//   phi[b,t] = sum_m alpha_m * exp(-beta_m * (kappa_m - t)^2), masked,
//   normalized by (row-sum + 1e-8). phi kept resident in LDS (16 KB)
//   so normalization never re-reads HBM. next_kappa emitted by lanes < 32.
// ---------------------------------------------------------------------------
__global__ __launch_bounds__(256)
void gmm_phi(const float* __restrict__ params,
             const float* __restrict__ prev_kappa,
             const int*   __restrict__ seqlen,
             float* __restrict__ out)
{
    __shared__ float s_alpha[M_MIX];
    __shared__ float s_beta [M_MIX];
    __shared__ float s_kappa[M_MIX];
    __shared__ float s_phi  [T_MAX];
    __shared__ float s_red  [256];

    const int b   = blockIdx.x;
    const int tid = threadIdx.x;

    if (tid < M_MIX) {
        const float ah = params[b * N3M +             tid];
        const float bh = params[b * N3M +     M_MIX + tid];
        const float kh = params[b * N3M + 2 * M_MIX + tid];
        const float alpha = __expf(ah);
        const float beta  = __expf(bh);
        const float kap   = prev_kappa[b * M_MIX + tid] + __expf(kh);
        s_alpha[tid] = alpha;
        s_beta [tid] = beta;
        s_kappa[tid] = kap;
        out[B_SZ * T_MAX + b * M_MIX + tid] = kap;   // next_kappa (2nd output)
    }
    __syncthreads();

    const int len = seqlen[b];
    float lsum = 0.0f;
    for (int t = tid; t < T_MAX; t += 256) {
        const float mu = (float)t;
        float phi = 0.0f;
        #pragma unroll
        for (int m = 0; m < M_MIX; ++m) {
            const float d = s_kappa[m] - mu;
            phi += s_alpha[m] * __expf(-s_beta[m] * d * d);
        }
        phi = (t < len) ? phi : 0.0f;
        s_phi[t] = phi;
        lsum += phi;
    }

    s_red[tid] = lsum;
    __syncthreads();
    #pragma unroll
    for (int off = 128; off > 0; off >>= 1) {
        if (tid < off) s_red[tid] += s_red[tid + off];
        __syncthreads();
    }
    const float inv = 1.0f / (s_red[0] + 1e-8f);

    for (int t = tid; t < T_MAX; t += 256) {
        out[b * T_MAX + t] = s_phi[t] * inv;
    }
}

// ---------------------------------------------------------------------------
extern "C" void kernel_launch(void* const* d_in, const int* in_sizes, int n_in,
                              void* d_out, int out_size, void* d_ws, size_t ws_size,
                              hipStream_t stream)
{
    const float* query      = (const float*)d_in[0];   // [512,1024]
    const float* prev_kappa = (const float*)d_in[1];   // [512,32]
    const float* W          = (const float*)d_in[2];   // [1024,96]
    const float* bias       = (const float*)d_in[3];   // [96]
    const int*   seqlen     = (const int*)  d_in[4];   // [512]
    float*       out        = (float*)d_out;           // [512*4096 + 512*32]
    float*       params     = (float*)d_ws;            // [512,96] scratch

    dim3 gridA(N3M / 16, B_SZ / 16);                   // (6, 32) waves
    gmm_proj_wmma<<<gridA, 32, 0, stream>>>(query, W, bias, params);

    gmm_phi<<<B_SZ, 256, 0, stream>>>(params, prev_kappa, seqlen, out);
}